// MutationGNN_87574383165811
// MI455X (gfx1250) — compile-verified
//
#include <hip/hip_runtime.h>
#include <hip/hip_bf16.h>

typedef __attribute__((ext_vector_type(2))) float v2f;
typedef __attribute__((ext_vector_type(8))) float v8f;

#define LDA 132   // 16x128 A tile padded: stride%64==4 -> conflict-free frag reads

__device__ __forceinline__ float relu_f(float v) { return v > 0.0f ? v : 0.0f; }

// ---------------- degree / norm kernels ----------------
__global__ void deg_init_kernel(float* __restrict__ deg, int n) {
    int i = blockIdx.x * blockDim.x + threadIdx.x;
    if (i < n) deg[i] = 1.0f;  // self loop
}

__global__ void deg_count_kernel(const int* __restrict__ dst, float* __restrict__ deg, int e) {
    int i = blockIdx.x * blockDim.x + threadIdx.x;
    if (i < e) atomicAdd(&deg[dst[i]], 1.0f);
}

__global__ void deg_to_dis_kernel(float* __restrict__ deg, int n) {
    int i = blockIdx.x * blockDim.x + threadIdx.x;
    if (i < n) deg[i] = rsqrtf(deg[i]);  // deg >= 1 always (self loop)
}

// ---------------- dense GEMM: H[Mx128] = act(A (+bias)) @ W[128x128] ----------------
// One block = 8 wave32s = one 16-row tile; wave w computes N-tile w (16 cols).
// A tile staged in LDS (bias+relu of previous layer fused into the staging load).
__global__ __launch_bounds__(256)
void gcn_gemm128_kernel(const float* __restrict__ A, const float* __restrict__ W,
                        const float* __restrict__ bias, float* __restrict__ H) {
    __shared__ float sA[16 * LDA];
    const int tid  = threadIdx.x;
    const int wave = tid >> 5;
    const int lane = tid & 31;
    const long row0 = (long)blockIdx.x * 16;

    // stage 16x128 A tile (2048 elems, 8 per thread), optional fused bias+relu
    for (int i = tid; i < 16 * 128; i += 256) {
        const int r = i >> 7, c = i & 127;
        float v = A[(row0 + r) * 128 + c];
        if (bias) v = relu_f(v + bias[c]);
        sA[r * LDA + c] = v;
    }
    __syncthreads();

    // f32 16x16x4 WMMA fragment addressing (wave32):
    //  A: lanes 0-15 rows 0..15 with K={k0,k0+1}; lanes 16-31 same rows K={k0+2,k0+3}
    //  B: VGPR0 = row k0 (lanes 0-15) / k0+2 (lanes 16-31); VGPR1 = k0+1 / k0+3
    const int arow = lane & 15;
    const int koff = (lane >> 4) << 1;          // 0 or 2
    const int col  = wave * 16 + (lane & 15);   // global output column

    v8f acc = {};
    #pragma unroll
    for (int kk = 0; kk < 32; ++kk) {
        const int ka = kk * 4 + koff;
        v2f a, b;
        a.x = sA[arow * LDA + ka];
        a.y = sA[arow * LDA + ka + 1];
        b.x = W[(ka + 0) * 128 + col];
        b.y = W[(ka + 1) * 128 + col];
        acc = __builtin_amdgcn_wmma_f32_16x16x4_f32(
            false, a, false, b, (short)0, acc, false, false);
    }

    // D layout: lanes 0-15 -> rows row0+v (v=0..7), lanes 16-31 -> rows row0+8+v
    const int rbase = (lane >> 4) * 8;
    #pragma unroll
    for (int v = 0; v < 8; ++v)
        H[(row0 + rbase + v) * 128 + col] = acc[v];
}

// ---------------- aggregation: out[i] = h[i]*dis[i]^2 ; out[dst] += h[src]*dis[src]*dis[dst] ----
__global__ void self_loop_init_kernel(const float* __restrict__ h, const float* __restrict__ dis,
                                      float* __restrict__ out, long total_vec4) {
    long i = (long)blockIdx.x * blockDim.x + threadIdx.x;  // over N*32 float4s
    if (i < total_vec4) {
        const int node = (int)(i >> 5);
        const float d = dis[node];
        const float nrm = d * d;
        float4 v = ((const float4*)h)[i];
        v.x *= nrm; v.y *= nrm; v.z *= nrm; v.w *= nrm;
        ((float4*)out)[i] = v;
    }
}

__global__ __launch_bounds__(256)
void edge_scatter_kernel(const int* __restrict__ src, const int* __restrict__ dst,
                         const float* __restrict__ h, const float* __restrict__ dis,
                         float* __restrict__ out, int e) {
    const long edge = (long)blockIdx.x * 8 + (threadIdx.x >> 5);  // one wave per edge
    const int lane = threadIdx.x & 31;                            // 4 channels per lane
    if (edge < e) {
        const int s = src[edge];
        const int d = dst[edge];
        const float nrm = dis[s] * dis[d];
        const float4 v = ((const float4*)(h + (long)s * 128))[lane];
        float* o = out + (long)d * 128 + lane * 4;
        atomicAdd(o + 0, v.x * nrm);
        atomicAdd(o + 1, v.y * nrm);
        atomicAdd(o + 2, v.z * nrm);
        atomicAdd(o + 3, v.w * nrm);
    }
}

// ---------------- final FC: out[Nx4] = relu(A + b2) @ Wfc[128x4] + bfc ----------------
// WMMA with B zero-padded to 16 columns; one wave per 16-row tile, 8 tiles per block.
__global__ __launch_bounds__(256)
void gcn_fc_kernel(const float* __restrict__ A, const float* __restrict__ bias_in,
                   const float* __restrict__ Wfc, const float* __restrict__ bfc,
                   float* __restrict__ out, int mtiles) {
    const int wave = threadIdx.x >> 5;
    const int lane = threadIdx.x & 31;
    const int tile = blockIdx.x * 8 + wave;
    if (tile >= mtiles) return;  // wave-uniform: EXEC stays all-ones for active waves

    const long row0 = (long)tile * 16;
    const int arow = lane & 15;
    const int koff = (lane >> 4) << 1;
    const int col  = lane & 15;

    v8f acc = {};
    #pragma unroll
    for (int kk = 0; kk < 32; ++kk) {
        const int ka = kk * 4 + koff;
        v2f a, b;
        a.x = relu_f(A[(row0 + arow) * 128 + ka]     + bias_in[ka]);
        a.y = relu_f(A[(row0 + arow) * 128 + ka + 1] + bias_in[ka + 1]);
        b.x = (col < 4) ? Wfc[(ka + 0) * 4 + col] : 0.0f;
        b.y = (col < 4) ? Wfc[(ka + 1) * 4 + col] : 0.0f;
        acc = __builtin_amdgcn_wmma_f32_16x16x4_f32(
            false, a, false, b, (short)0, acc, false, false);
    }

    if (col < 4) {
        const int rbase = (lane >> 4) * 8;
        const float bb = bfc[col];
        #pragma unroll
        for (int v = 0; v < 8; ++v)
            out[(row0 + rbase + v) * 4 + col] = acc[v] + bb;
    }
}

// ---------------- launcher ----------------
extern "C" void kernel_launch(void* const* d_in, const int* in_sizes, int n_in,
                              void* d_out, int out_size, void* d_ws, size_t ws_size,
                              hipStream_t stream) {
    const float* x   = (const float*)d_in[0];
    const int*   ei  = (const int*)d_in[1];   // [2, E]: src then dst
    const float* W1  = (const float*)d_in[2];
    const float* b1  = (const float*)d_in[3];
    const float* W2  = (const float*)d_in[4];
    const float* b2  = (const float*)d_in[5];
    const float* Wfc = (const float*)d_in[6];
    const float* bfc = (const float*)d_in[7];
    float* out = (float*)d_out;

    const int N = in_sizes[0] / 128;  // 50000
    const int E = in_sizes[1] / 2;    // 800000
    const int mtiles = N / 16;        // 3125 (exact)
    const int* src = ei;
    const int* dst = ei + E;

    // workspace: dis[N] | bufA[N*128] | bufB[N*128]
    float* dis  = (float*)d_ws;
    float* bufA = dis + (((size_t)N + 255) & ~(size_t)255);
    float* bufB = bufA + (size_t)N * 128;

    const int T = 256;
    // 1) deg -> deg_inv_sqrt (in dis)
    deg_init_kernel<<<(N + T - 1) / T, T, 0, stream>>>(dis, N);
    deg_count_kernel<<<(E + T - 1) / T, T, 0, stream>>>(dst, dis, E);
    deg_to_dis_kernel<<<(N + T - 1) / T, T, 0, stream>>>(dis, N);

    const long v4 = (long)N * 32;              // N*128 floats as float4
    const int selfBlocks = (int)((v4 + T - 1) / T);
    const int edgeBlocks = (E + 7) / 8;        // one wave per edge

    // 2) layer 1: h1 = x @ W1 ; agg1 = A_hat h1
    gcn_gemm128_kernel<<<mtiles, T, 0, stream>>>(x, W1, nullptr, bufA);
    self_loop_init_kernel<<<selfBlocks, T, 0, stream>>>(bufA, dis, bufB, v4);
    edge_scatter_kernel<<<edgeBlocks, T, 0, stream>>>(src, dst, bufA, dis, bufB, E);

    // 3) layer 2: h2 = relu(agg1 + b1) @ W2 ; agg2 = A_hat h2
    gcn_gemm128_kernel<<<mtiles, T, 0, stream>>>(bufB, W2, b1, bufA);
    self_loop_init_kernel<<<selfBlocks, T, 0, stream>>>(bufA, dis, bufB, v4);
    edge_scatter_kernel<<<edgeBlocks, T, 0, stream>>>(src, dst, bufA, dis, bufB, E);

    // 4) out = relu(agg2 + b2) @ Wfc + bfc
    gcn_fc_kernel<<<(mtiles + 7) / 8, T, 0, stream>>>(bufB, b2, Wfc, bfc, out, mtiles);
}